// PointConv_68745246539912
// MI455X (gfx1250) — compile-verified
//
#include <hip/hip_runtime.h>
#include <hip/hip_bf16.h>

// ---------------------------------------------------------------------------
// PointConv-style edge aggregation, restructured to exploit linearity:
//   accum[q, 0:128]  += ref_feat[e_ref]          (per edge)
//   accum[q,128:131] += pos_diff(e)              (per edge)
//   cnt[q]           += 1
//   qf = (accum @ Wcat + cnt*bias) / max(cnt,1)  (fp32 WMMA 16x16x4)
//   out = ReLU(BN(qf))                            (batch stats)
// ---------------------------------------------------------------------------

typedef __attribute__((ext_vector_type(2))) float v2f;
typedef __attribute__((ext_vector_type(8))) float v8f;

#define KTOT 132   // 128 feat + 3 pos + 1 pad
#define COUT 128

// ---------------- kernel 1: zero the atomic accumulators --------------------
__global__ void zero_ws_kernel(float* __restrict__ p, size_t n) {
    size_t i = (size_t)blockIdx.x * blockDim.x + threadIdx.x;
    size_t stride = (size_t)gridDim.x * blockDim.x;
    for (; i < n; i += stride) p[i] = 0.0f;
}

// ---------------- kernel 2: pack concatenated weight / bias -----------------
// Wcat[k][c]: k<128 -> W_mlp[c][k]; k=128..130 -> W_pos[c][k-128]; k=131 -> 0
__global__ void pack_weights_kernel(const float* __restrict__ W_pos,
                                    const float* __restrict__ b_pos,
                                    const float* __restrict__ W_mlp,
                                    const float* __restrict__ b_mlp,
                                    float* __restrict__ Wcat,
                                    float* __restrict__ biascat) {
    int c = blockIdx.x * blockDim.x + threadIdx.x;
    if (c < COUT) {
        for (int k = 0; k < 128; ++k) Wcat[k * COUT + c] = W_mlp[c * 128 + k];
        for (int k = 0; k < 3; ++k)   Wcat[(128 + k) * COUT + c] = W_pos[c * 3 + k];
        Wcat[131 * COUT + c] = 0.0f;
        biascat[c] = b_pos[c] + b_mlp[c];
    }
}

// ---------------- kernel 3: edge scatter (one wave32 per edge) --------------
// lane l handles channels [4l, 4l+4): coalesced 16B gather from ref_feat
// (51 MB table -> L2-resident), fp32 hardware atomics into accum (26 MB, L2).
__global__ void edge_scatter_kernel(const float* __restrict__ ref_bxyz,
                                    const float* __restrict__ ref_feat,
                                    const float* __restrict__ query_bxyz,
                                    const int* __restrict__ e_ref,
                                    const int* __restrict__ e_query,
                                    float* __restrict__ accum,   // [M x 132]
                                    float* __restrict__ cnt,     // [M]
                                    int E) {
    int tid  = blockIdx.x * blockDim.x + threadIdx.x;
    int edge = tid >> 5;
    int lane = tid & 31;
    if (edge >= E) return;

    int r = e_ref[edge];
    int q = e_query[edge];

    const float4 f = *(const float4*)(ref_feat + (size_t)r * 128 + lane * 4);
    float* dst = accum + (size_t)q * KTOT + lane * 4;
    unsafeAtomicAdd(dst + 0, f.x);
    unsafeAtomicAdd(dst + 1, f.y);
    unsafeAtomicAdd(dst + 2, f.z);
    unsafeAtomicAdd(dst + 3, f.w);

    if (lane == 0) {
        const float* rb = ref_bxyz   + (size_t)r * 4;
        const float* qb = query_bxyz + (size_t)q * 4;
        float* pdst = accum + (size_t)q * KTOT + 128;
        unsafeAtomicAdd(pdst + 0, rb[1] - qb[1]);
        unsafeAtomicAdd(pdst + 1, rb[2] - qb[2]);
        unsafeAtomicAdd(pdst + 2, rb[3] - qb[3]);
        unsafeAtomicAdd(cnt + q, 1.0f);
    }
}

// ---------------- kernel 4: fused GEMM (fp32 WMMA) + mean + BN stats --------
// Block = 256 threads = 8 waves; block covers 16 query rows x 128 channels;
// each wave computes one 16x16 tile via V_WMMA_F32_16X16X4_F32, K=132.
__global__ void gemm_mean_stats_kernel(const float* __restrict__ X,      // [M x 132]
                                       const float* __restrict__ Wcat,   // [132 x 128]
                                       const float* __restrict__ biascat,// [128]
                                       const float* __restrict__ cnt,    // [M]
                                       float* __restrict__ out,          // [M x 128]
                                       float* __restrict__ stats,        // [256] sum|sumsq
                                       int Mq) {
    __shared__ float s_sum[COUT];
    __shared__ float s_sq[COUT];
    int tid = threadIdx.x;
    if (tid < COUT) { s_sum[tid] = 0.0f; s_sq[tid] = 0.0f; }
    __syncthreads();

    const int wave  = tid >> 5;
    const int lane  = tid & 31;
    const int lhalf = lane >> 4;   // 0: K+{0,1} / rows M..M+7 ; 1: K+{2,3} / rows M+8..
    const int lmod  = lane & 15;

    const int row0 = blockIdx.x * 16;
    const int col0 = wave * 16;

    // Clamp A-row for loads so EXEC stays all-ones through the WMMAs.
    int arow_i = row0 + lmod;
    if (arow_i >= Mq) arow_i = Mq - 1;
    const float* arow = X + (size_t)arow_i * KTOT;

    v8f acc = {};
    for (int k = 0; k < KTOT; k += 4) {
        // A 16x4 fp32: lane holds K = k+2*lhalf, k+2*lhalf+1 for row lmod
        v2f a = *(const v2f*)(arow + k + 2 * lhalf);
        // B 4x16 fp32: lane holds Wcat[k+2*lhalf .. +1][col0+lmod]
        v2f b;
        b.x = Wcat[(size_t)(k + 2 * lhalf)     * COUT + col0 + lmod];
        b.y = Wcat[(size_t)(k + 2 * lhalf + 1) * COUT + col0 + lmod];
        acc = __builtin_amdgcn_wmma_f32_16x16x4_f32(
            /*neg_a=*/false, a, /*neg_b=*/false, b,
            /*c_mod=*/(short)0, acc, /*reuse_a=*/false, /*reuse_b=*/false);
    }

    // Epilogue: D tile layout -> VGPR r holds row (row0 + r + 8*lhalf), col = col0+lmod
    const int col  = col0 + lmod;
    const float bias = biascat[col];
    float lsum = 0.0f, lsq = 0.0f;
    #pragma unroll
    for (int r = 0; r < 8; ++r) {
        int row = row0 + r + 8 * lhalf;
        if (row < Mq) {
            float c   = cnt[row];
            float inv = 1.0f / fmaxf(c, 1.0f);
            float v   = (acc[r] + bias * c) * inv;   // scatter-mean incl. bias
            out[(size_t)row * COUT + col] = v;
            lsum += v;
            lsq  += v * v;
        }
    }
    // per-channel BN partial sums: LDS reduce, then one global atomic per chan
    unsafeAtomicAdd(&s_sum[col], lsum);
    unsafeAtomicAdd(&s_sq[col],  lsq);
    __syncthreads();
    if (tid < COUT) {
        unsafeAtomicAdd(&stats[tid],        s_sum[tid]);
        unsafeAtomicAdd(&stats[COUT + tid], s_sq[tid]);
    }
}

// ---------------- kernel 5: BatchNorm (batch stats) + ReLU, in place --------
__global__ void bn_relu_kernel(float* __restrict__ out,
                               const float* __restrict__ stats,
                               const float* __restrict__ gamma,
                               const float* __restrict__ beta,
                               int Mq) {
    size_t total = (size_t)Mq * COUT;
    size_t i = (size_t)blockIdx.x * blockDim.x + threadIdx.x;
    size_t stride = (size_t)gridDim.x * blockDim.x;
    float invM = 1.0f / (float)Mq;
    for (; i < total; i += stride) {
        int c    = (int)(i & (COUT - 1));
        float mu = stats[c] * invM;
        float var = stats[COUT + c] * invM - mu * mu;
        float sc = gamma[c] * rsqrtf(var + 1e-5f);
        float v  = (out[i] - mu) * sc + beta[c];
        out[i] = fmaxf(v, 0.0f);
    }
}

// ---------------------------------------------------------------------------
extern "C" void kernel_launch(void* const* d_in, const int* in_sizes, int n_in,
                              void* d_out, int out_size, void* d_ws, size_t ws_size,
                              hipStream_t stream) {
    const float* ref_bxyz   = (const float*)d_in[0];
    const float* ref_feat   = (const float*)d_in[1];
    const float* query_bxyz = (const float*)d_in[2];
    const int*   e_ref      = (const int*)d_in[3];
    const int*   e_query    = (const int*)d_in[4];
    const float* W_pos      = (const float*)d_in[5];
    const float* b_pos      = (const float*)d_in[6];
    const float* W_mlp      = (const float*)d_in[7];
    const float* b_mlp      = (const float*)d_in[8];
    const float* gamma      = (const float*)d_in[9];
    const float* beta       = (const float*)d_in[10];
    float* out = (float*)d_out;

    const int Mq = in_sizes[2] / 4;   // query_bxyz: [M,4]
    const int E  = in_sizes[3];       // e_ref: [E]

    // workspace layout (floats):
    //   [accum M*132][cnt M][stats 256][Wcat 132*128][biascat 128]
    float* accum   = (float*)d_ws;
    float* cnt     = accum + (size_t)Mq * KTOT;
    float* stats   = cnt + Mq;
    float* Wcat    = stats + 256;
    float* biascat = Wcat + (size_t)KTOT * COUT;

    // 1) zero the atomic accumulators (accum + cnt + stats are contiguous)
    size_t nzero = (size_t)Mq * KTOT + (size_t)Mq + 256;
    zero_ws_kernel<<<4096, 256, 0, stream>>>(accum, nzero);

    // 2) pack concatenated weights
    pack_weights_kernel<<<1, 128, 0, stream>>>(W_pos, b_pos, W_mlp, b_mlp,
                                               Wcat, biascat);

    // 3) edge scatter: one wave32 per edge
    {
        long long threads = (long long)E * 32;
        int blocks = (int)((threads + 255) / 256);
        edge_scatter_kernel<<<blocks, 256, 0, stream>>>(
            ref_bxyz, ref_feat, query_bxyz, e_ref, e_query, accum, cnt, E);
    }

    // 4) fused GEMM (fp32 WMMA) + scatter-mean epilogue + BN stat reduction
    {
        int blocks = (Mq + 15) / 16;
        gemm_mean_stats_kernel<<<blocks, 256, 0, stream>>>(
            accum, Wcat, biascat, cnt, out, stats, Mq);
    }

    // 5) BatchNorm + ReLU in place on d_out
    bn_relu_kernel<<<2048, 256, 0, stream>>>(out, stats, gamma, beta, Mq);
}